// MultiHeadSpectralAttention_22832046146362
// MI455X (gfx1250) — compile-verified
//
#include <hip/hip_runtime.h>
#include <math.h>

// ---------------------------------------------------------------------------
// MultiHeadSpectralAttention forward for MI455X (gfx1250, wave32)
//   x:[16,2048,768] f32, FFT(N=2048) per (b,h,c) column, adaptive filter from
//   mean-pooled context MLP (WMMA f32 16x16x4), complex GELU, IFFT, residual+LN.
// Memory-bound (~0.5 GB traffic vs a few GFLOP) -> fp32 throughout,
// transpose-to-contiguous-columns layout, FFT fully in LDS, in-place buffers.
// Workspace use: xt (100.7MB) + ctx/h1/mod (~1.6MB) ~= 102.4 MB.
// ---------------------------------------------------------------------------

typedef __attribute__((ext_vector_type(2))) float v2f;
typedef __attribute__((ext_vector_type(8))) float v8f;

#define BSZ   16
#define NN    2048
#define DD    768
#define HH    12
#define HD    64
#define LOG2N 11
#define PI_F  3.14159265358979323846f

__device__ __forceinline__ float gelu_exact(float v) {
    // torch nn.GELU default (erf form)
    return 0.5f * v * (1.0f + erff(v * 0.70710678118654752440f));
}

// ------------------------- context = mean_n x[b,n,d] ------------------------
__global__ void context_mean_kernel(const float* __restrict__ x,
                                    float* __restrict__ ctx) {
    const int b  = blockIdx.x / 3;
    const int d  = (blockIdx.x % 3) * 256 + threadIdx.x;   // coalesced in d
    const float* xp = x + (size_t)b * NN * DD + d;
    float acc = 0.0f;
    for (int n = 0; n < NN; ++n) acc += xp[(size_t)n * DD];
    ctx[b * DD + d] = acc * (1.0f / (float)NN);
}

// ----------------- WMMA f32 GEMM: Out[16,Nn] = act(A[16,K]*B + bias) --------
// One wave computes one 16x16 tile with V_WMMA_F32_16X16X4_F32 (K-step 4).
// A staged in LDS, row pitch 772 (768+4) so the 16-lane column read hits
// banks 4*lm+k (conflict-free; 768 % 64 == 0 would be a 16-way conflict).
#define A_PITCH 772
__global__ void wmma_gemm_bias_act(const float* __restrict__ A,   // [16,768]
                                   const float* __restrict__ Bm,  // [768,Nn]
                                   const float* __restrict__ bias,// [Nn]
                                   float* __restrict__ Out,       // [16,Nn]
                                   int Nn, int do_gelu) {
    __shared__ float As[16 * A_PITCH];
    const int tid = threadIdx.x;
    for (int i = tid; i < 16 * DD; i += blockDim.x) {
        const int r = i / DD, c = i - r * DD;
        As[r * A_PITCH + c] = A[i];
    }
    __syncthreads();

    const int wave = tid >> 5;
    const int lane = tid & 31;
    const int n0   = (blockIdx.x * 8 + wave) * 16;
    if (n0 >= Nn) return;

    const int lh = lane >> 4;   // half: K offset 0 or 2 (A), rows 0-7/8-15 (D)
    const int lm = lane & 15;

    v8f acc = {};
    for (int k0 = 0; k0 < DD; k0 += 4) {
        const int ka = k0 + 2 * lh;
        v2f av = { As[lm * A_PITCH + ka], As[lm * A_PITCH + ka + 1] };
        const float* bp = Bm + (size_t)ka * Nn + n0 + lm;
        v2f bv = { bp[0], bp[Nn] };
        acc = __builtin_amdgcn_wmma_f32_16x16x4_f32(
            false, av, false, bv, (short)0, acc, false, false);
    }

    const int n = n0 + lm;
    const float bb = bias[n];
    #pragma unroll
    for (int r = 0; r < 8; ++r) {
        const int mrow = r + 8 * lh;          // C/D layout: lanes 16-31 -> M+8
        float v = acc[r] + bb;
        if (do_gelu) v = gelu_exact(v);
        Out[(size_t)mrow * Nn + n] = v;
    }
}

// ----------------- transpose x[b,n,h*64+c] -> xt[(b*H+h)*64+c][n] -----------
__global__ void transpose_fwd_kernel(const float* __restrict__ x,
                                     float* __restrict__ xt) {
    __shared__ float tile[64][65];
    const int blk = blockIdx.x;
    const int nt  = blk & 31;           // 32 tiles of 64 tokens
    const int bh  = blk >> 5;           // b*H + h
    const int h   = bh % HH;
    const int b   = bh / HH;
    const int n0  = nt * 64;
    const int tr  = threadIdx.x >> 6;   // 0..3
    const int tc  = threadIdx.x & 63;
    #pragma unroll
    for (int i = 0; i < 16; ++i) {      // coalesced read: 64 floats per row
        const int r = i * 4 + tr;
        tile[r][tc] = x[((size_t)(b * NN + n0 + r)) * DD + h * HD + tc];
    }
    __syncthreads();
    #pragma unroll
    for (int i = 0; i < 16; ++i) {      // coalesced write along n
        const int cc = i * 4 + tr;
        xt[((size_t)(bh * HD + cc)) * NN + n0 + tc] = tile[tc][cc];
    }
}

// -------- transpose xt (now holding filtered signal) -> out[b,n,d] ----------
__global__ void transpose_bwd_kernel(const float* __restrict__ xt,
                                     float* __restrict__ xf) {
    __shared__ float tile[64][65];
    const int blk = blockIdx.x;
    const int nt  = blk & 31;
    const int bh  = blk >> 5;
    const int h   = bh % HH;
    const int b   = bh / HH;
    const int n0  = nt * 64;
    const int tr  = threadIdx.x >> 6;
    const int tc  = threadIdx.x & 63;
    #pragma unroll
    for (int i = 0; i < 16; ++i) {
        const int cc = i * 4 + tr;
        tile[cc][tc] = xt[((size_t)(bh * HD + cc)) * NN + n0 + tc];
    }
    __syncthreads();
    #pragma unroll
    for (int i = 0; i < 16; ++i) {
        const int r = i * 4 + tr;
        xf[((size_t)(b * NN + n0 + r)) * DD + h * HD + tc] = tile[tc][r];
    }
}

// ----------------- radix-2 Stockham (autosort, ping-pong in LDS) ------------
__device__ __forceinline__ void fft2048(float*& sr, float*& si,
                                        float*& dr, float*& di,
                                        float sign, int tid) {
    int l = NN / 2;                       // 1024
    for (int stage = 0; stage < LOG2N; ++stage) {
        const int   m  = 1 << stage;
        const float as = sign * PI_F / (float)l;   // angle = as * j
        for (int idx = tid; idx < NN / 2; idx += 256) {
            const int j  = idx >> stage;           // idx = j*m + k
            const int o0 = idx + (j << stage);     // k + 2*j*m
            float wi_, wr_;
            __sincosf(as * (float)j, &wi_, &wr_);
            const float c0r = sr[idx],            c0i = si[idx];
            const float c1r = sr[idx + NN / 2],   c1i = si[idx + NN / 2];
            dr[o0]     = c0r + c1r;
            di[o0]     = c0i + c1i;
            const float tr_ = c0r - c1r, ti_ = c0i - c1i;
            dr[o0 + m] = tr_ * wr_ - ti_ * wi_;
            di[o0 + m] = tr_ * wi_ + ti_ * wr_;
        }
        __syncthreads();
        float* t;
        t = sr; sr = dr; dr = t;
        t = si; si = di; di = t;
        l >>= 1;
    }
}

// ----- per-column: FFT -> (base_filter+mod) scale -> GELU(re),GELU(im) ------
// ----- -> IFFT -> real/N, written back in place into xt ---------------------
__global__ void fft_filter_ifft_kernel(float* __restrict__ xt,
                                       const float* __restrict__ mod,  // [B,H*N]
                                       const float* __restrict__ bf) { // [H,N]
    __shared__ float ar[NN], ai[NN], br_[NN], bi_[NN];
    const int col = blockIdx.x;            // (b*H + h)*64 + c
    const int bh  = col >> 6;
    const int h   = bh % HH;
    const int b   = bh / HH;
    float* base = xt + (size_t)col * NN;
    const int tid = threadIdx.x;           // 256 threads

    for (int i = tid; i < NN; i += 256) { ar[i] = base[i]; ai[i] = 0.0f; }
    __syncthreads();

    float *sr = ar, *si = ai, *dr = br_, *di = bi_;
    fft2048(sr, si, dr, di, -1.0f, tid);   // forward

    const float* mp = mod + ((size_t)b * HH + h) * NN;
    const float* fp = bf + (size_t)h * NN;
    for (int k = tid; k < NN; k += 256) {
        const float f = fp[k] + mp[k];
        sr[k] = gelu_exact(sr[k] * f);
        si[k] = gelu_exact(si[k] * f);
    }
    __syncthreads();

    fft2048(sr, si, dr, di, 1.0f, tid);    // inverse (unscaled)

    for (int i = tid; i < NN; i += 256)
        base[i] = sr[i] * (1.0f / (float)NN);  // real part of ifft
}

// ----------------- y = LN(x + xf) over D, in place on d_out -----------------
__device__ __forceinline__ float wave_reduce_add(float v) {
    #pragma unroll
    for (int off = 16; off > 0; off >>= 1) v += __shfl_down(v, off, 32);
    return v;
}

__global__ void residual_ln_kernel(const float* __restrict__ x,
                                   float* __restrict__ out,   // xf in, y out
                                   const float* __restrict__ g,
                                   const float* __restrict__ be) {
    const int row = blockIdx.x;            // b*N + n
    const float* xr = x + (size_t)row * DD;
    float* yr = out + (size_t)row * DD;
    const int tid = threadIdx.x;           // 256 -> 3 elems/thread

    float vals[3];
    float s = 0.0f, ss = 0.0f;
    #pragma unroll
    for (int j = 0; j < 3; ++j) {
        const int d = tid + j * 256;
        const float v = xr[d] + yr[d];
        vals[j] = v; s += v; ss += v * v;
    }
    __shared__ float sbuf[8], ssbuf[8];
    s  = wave_reduce_add(s);
    ss = wave_reduce_add(ss);
    const int wave = tid >> 5, lane = tid & 31;
    if (lane == 0) { sbuf[wave] = s; ssbuf[wave] = ss; }
    __syncthreads();
    if (wave == 0) {
        float a = (lane < 8) ? sbuf[lane]  : 0.0f;
        float c = (lane < 8) ? ssbuf[lane] : 0.0f;
        a = wave_reduce_add(a);
        c = wave_reduce_add(c);
        if (lane == 0) { sbuf[0] = a; ssbuf[0] = c; }
    }
    __syncthreads();
    const float mean = sbuf[0] * (1.0f / (float)DD);
    const float var  = ssbuf[0] * (1.0f / (float)DD) - mean * mean;
    const float inv  = rsqrtf(var + 1e-5f);
    #pragma unroll
    for (int j = 0; j < 3; ++j) {
        const int d = tid + j * 256;
        yr[d] = (vals[j] - mean) * inv * g[d] + be[d];
    }
}

// ---------------------------------------------------------------------------
extern "C" void kernel_launch(void* const* d_in, const int* in_sizes, int n_in,
                              void* d_out, int out_size, void* d_ws, size_t ws_size,
                              hipStream_t stream) {
    const float* x  = (const float*)d_in[0];   // [16,2048,768]
    const float* bf = (const float*)d_in[1];   // [12,2048,1]
    const float* w1 = (const float*)d_in[2];   // [768,768]
    const float* b1 = (const float*)d_in[3];   // [768]
    const float* w2 = (const float*)d_in[4];   // [768,24576]
    const float* b2 = (const float*)d_in[5];   // [24576]
    const float* lg = (const float*)d_in[6];   // [768]
    const float* lb = (const float*)d_in[7];   // [768]
    float* out = (float*)d_out;                // [16,2048,768]
    float* ws  = (float*)d_ws;

    float* xt  = ws;                            // 25,165,824 f (100.7 MB)
    float* ctx = xt + (size_t)BSZ * HH * HD * NN;
    float* h1  = ctx + BSZ * DD;
    float* mod = h1 + BSZ * DD;                 // 393,216 f

    // context MLP path (tiny, WMMA fp32)
    context_mean_kernel<<<BSZ * 3, 256, 0, stream>>>(x, ctx);
    wmma_gemm_bias_act<<<6,   256, 0, stream>>>(ctx, w1, b1, h1, DD, 1);
    wmma_gemm_bias_act<<<192, 256, 0, stream>>>(h1, w2, b2, mod, HH * NN, 0);

    // spectral path
    transpose_fwd_kernel<<<BSZ * HH * 32, 256, 0, stream>>>(x, xt);
    fft_filter_ifft_kernel<<<BSZ * HH * HD, 256, 0, stream>>>(xt, mod, bf);
    transpose_bwd_kernel<<<BSZ * HH * 32, 256, 0, stream>>>(xt, out);

    // residual + LayerNorm (in place on d_out)
    residual_ln_kernel<<<BSZ * NN, 256, 0, stream>>>(x, out, lg, lb);
}